// GCNResnet_10823317585955
// MI455X (gfx1250) — compile-verified
//
#include <hip/hip_runtime.h>
#include <hip/hip_bf16.h>
#include <math.h>

// ---------------------------------------------------------------------------
// Problem dimensions (fixed by the reference)
// ---------------------------------------------------------------------------
#define BB    64      // batch
#define CC    80      // classes
#define DE    300
#define DE1   1024
#define JJ    4000
#define SS    100
#define KK    40      // JJ / SS
#define FDIM  2048

typedef __attribute__((ext_vector_type(2))) float v2f;
typedef __attribute__((ext_vector_type(8))) float v8f;

__device__ __forceinline__ v2f ldv2(const float* p) {
    return *reinterpret_cast<const v2f*>(p);
}

// ---------------------------------------------------------------------------
// Kernel 1: global max-pool over 14x14.  feature [B,2048,14,14] -> feat [B,2048]
// One wave32 per (b,c) slab of 196 contiguous floats: coalesced 128B bursts,
// then a 5-step cross-lane max (ds permute path).
// ---------------------------------------------------------------------------
__global__ __launch_bounds__(256)
void maxpool14_kernel(const float* __restrict__ feature, float* __restrict__ feat)
{
    int wave = (blockIdx.x * blockDim.x + threadIdx.x) >> 5;
    int lane = threadIdx.x & 31;
    if (wave >= BB * FDIM) return;
    const float* p = feature + (long)wave * 196;
    float v = -INFINITY;
    #pragma unroll
    for (int i = lane; i < 196; i += 32) v = fmaxf(v, p[i]);
    #pragma unroll
    for (int off = 16; off > 0; off >>= 1)
        v = fmaxf(v, __shfl_xor(v, off, 32));
    if (lane == 0) feat[wave] = v;
}

// ---------------------------------------------------------------------------
// Generic strided f32 WMMA GEMM with NT independent 16-wide column tiles per
// wave:  D[m,n] = act( alpha * sum_k A[m,k]*B[k,n] + bias )
//   A element (m,k) at A[m*rsA + k*csA];  B element (k,n) at B[k*rsB + n*csB]
//   One wave computes a 16 x (16*NT) strip: the A fragment is loaded once per
//   K-step and reused across NT V_WMMA_F32_16X16X4_F32 issues into NT
//   independent accumulators (WMMA ILP; no D->C stall chain).
// BIAS_MODE: 0 none, 1 bias[n], 2 bias[m].   ACT: 0 none, 1 sigmoid, 2 leaky(0.2)
// AVEC/BVEC: k-stride == 1 for that operand -> 8-byte vector loads.
// Requires M%16==0, N%(16*NT)==0, K%4==0 (true for every GEMM in this model).
// ---------------------------------------------------------------------------
template<int BIAS_MODE, int ACT, bool AVEC, bool BVEC, int NT>
__global__ __launch_bounds__(32)
void wmma_gemm_f32(const float* __restrict__ A, const float* __restrict__ B,
                   const float* __restrict__ bias, float* __restrict__ D,
                   int K, long rsA, long csA, long rsB, long csB, long ldd,
                   float alpha)
{
    const int lane = threadIdx.x;          // 0..31, EXEC all ones (WMMA req.)
    const int half = lane >> 4;            // 0 or 1
    const int l15  = lane & 15;
    const int m    = blockIdx.y * 16 + l15;            // A row for this lane
    const int n0   = blockIdx.x * (16 * NT) + l15;     // first B col for lane
    const int kb   = half * 2;                         // ISA A/B K sub-assignment

    const float* __restrict__ Arow = A + (long)m * rsA;
    const float* Bcol[NT];
    #pragma unroll
    for (int t = 0; t < NT; ++t)
        Bcol[t] = B + (long)(n0 + 16 * t) * csB;

    v8f acc[NT];
    #pragma unroll
    for (int t = 0; t < NT; ++t) acc[t] = (v8f){};

    #pragma unroll 2
    for (int k = 0; k < K; k += 4) {
        const int k0 = k + kb;
        v2f a;
        if (AVEC) {
            a = ldv2(Arow + k0);                       // global_load_b64
        } else {
            a.x = Arow[(long)(k0    ) * csA];
            a.y = Arow[(long)(k0 + 1) * csA];
        }
        v2f b[NT];
        #pragma unroll
        for (int t = 0; t < NT; ++t) {
            if (BVEC) {
                b[t] = ldv2(Bcol[t] + k0);             // global_load_b64
            } else {
                b[t].x = Bcol[t][(long)(k0    ) * rsB];
                b[t].y = Bcol[t][(long)(k0 + 1) * rsB];
            }
        }
        #pragma unroll
        for (int t = 0; t < NT; ++t)
            acc[t] = __builtin_amdgcn_wmma_f32_16x16x4_f32(
                         false, a, false, b[t], (short)0, acc[t], false, false);
    }

    // C/D layout: VGPR r, lanes 0-15 -> M=r ; lanes 16-31 -> M=r+8
    #pragma unroll
    for (int t = 0; t < NT; ++t) {
        const int n = n0 + 16 * t;
        #pragma unroll
        for (int r = 0; r < 8; ++r) {
            const int mm = blockIdx.y * 16 + r + half * 8;
            float v = acc[t][r] * alpha;
            if (BIAS_MODE == 1) v += bias[n];
            if (BIAS_MODE == 2) v += bias[mm];
            if (ACT == 1) v = 1.0f / (1.0f + __expf(-v));
            if (ACT == 2) v = (v > 0.0f) ? v : 0.2f * v;
            D[(long)mm * ldd + n] = v;
        }
    }
}

// ---------------------------------------------------------------------------
// Kernel: A_wave = Araw + I ; d = rowsum^-1/2 (non-finite -> 0) ;
// A_hat = d_i * A_wave * d_j ; L_A_loss = sum |A_hat - I|.  Single block.
// ---------------------------------------------------------------------------
__global__ __launch_bounds__(256)
void ahat_kernel(const float* __restrict__ Araw, float* __restrict__ Ahat,
                 float* __restrict__ loss_out)
{
    __shared__ float dvec[CC];
    __shared__ float red[256];
    const int t = threadIdx.x;

    if (t < CC) {
        float s = 0.0f;
        #pragma unroll 4
        for (int j = 0; j < CC; ++j)
            s += Araw[t * CC + j] + ((j == t) ? 1.0f : 0.0f);
        float dv = rsqrtf(s);                 // s^-0.5
        if (!isfinite(dv)) dv = 0.0f;
        dvec[t] = dv;
    }
    __syncthreads();

    float lacc = 0.0f;
    for (int idx = t; idx < CC * CC; idx += 256) {
        const int i = idx / CC, j = idx % CC;
        const float eye = (i == j) ? 1.0f : 0.0f;
        const float ah  = dvec[i] * (Araw[idx] + eye) * dvec[j];
        Ahat[idx] = ah;
        lacc += fabsf(ah - eye);
    }
    red[t] = lacc;
    __syncthreads();
    for (int s = 128; s > 0; s >>= 1) {
        if (t < s) red[t] += red[t + s];
        __syncthreads();
    }
    if (t == 0) loss_out[0] = red[0];
}

// ---------------------------------------------------------------------------
// Kernel: MFB sum-pool  pooled[b,c,k] = sum_{s<100} img[b,k*100+s]*cls[c,k*100+s]
// ---------------------------------------------------------------------------
__global__ __launch_bounds__(256)
void mfb_pool_kernel(const float* __restrict__ img, const float* __restrict__ cls,
                     float* __restrict__ pooled)
{
    const int idx = blockIdx.x * blockDim.x + threadIdx.x;   // b*3200 + c*40 + k
    if (idx >= BB * CC * KK) return;
    const int k = idx % KK;
    const int c = (idx / KK) % CC;
    const int b = idx / (CC * KK);
    const float* __restrict__ ip = img + (long)b * JJ + k * SS;
    const float* __restrict__ cp = cls + (long)c * JJ + k * SS;
    float s = 0.0f;
    #pragma unroll 4
    for (int t = 0; t < SS; ++t) s += ip[t] * cp[t];
    pooled[idx] = s;
}

// ---------------------------------------------------------------------------
// Host-side orchestration
// ---------------------------------------------------------------------------
extern "C" void kernel_launch(void* const* d_in, const int* in_sizes, int n_in,
                              void* d_out, int out_size, void* d_ws, size_t ws_size,
                              hipStream_t stream)
{
    const float* feature = (const float*)d_in[0];   // [64,2048,14,14]
    const float* inp     = (const float*)d_in[1];   // [80,300]
    const float* Wb1     = (const float*)d_in[2];   // [1024,300]
    const float* bb1     = (const float*)d_in[3];   // [1024]
    const float* Wb2     = (const float*)d_in[4];
    const float* bb2     = (const float*)d_in[5];
    const float* Wg1     = (const float*)d_in[6];   // [300,1024]
    const float* Wg2     = (const float*)d_in[7];   // [1024,2048]
    const float* Wimg    = (const float*)d_in[8];   // [4000,2048]
    const float* bimg    = (const float*)d_in[9];   // [4000]
    const float* Wcls    = (const float*)d_in[10];  // [4000,2048]
    const float* bcls    = (const float*)d_in[11];  // [4000]
    const float* Wml     = (const float*)d_in[12];  // [80,3200]
    const float* bml     = (const float*)d_in[13];  // [80]
    float* out = (float*)d_out;                     // [64*80] ++ [1] loss

    // workspace layout (floats), ~6.3 MB total
    float* ws = (float*)d_ws;
    float* feat   = ws;                 // 131072
    float* b1     = feat   + 131072;    //  81920
    float* b2     = b1     +  81920;    //  81920
    float* Araw   = b2     +  81920;    //   6400
    float* Ahat   = Araw   +   6400;    //   6400
    float* t1     = Ahat   +   6400;    //  81920
    float* h      = t1     +  81920;    //  81920
    float* t2     = h      +  81920;    // 163840
    float* x      = t2     + 163840;    // 163840
    float* img    = x      + 163840;    // 256000
    float* cls    = img    + 256000;    // 320000
    float* pooled = cls    + 320000;    // 204800

    // 1) max-pool: one wave per (b,c)
    {
        int waves = BB * FDIM;
        maxpool14_kernel<<<(waves * 32 + 255) / 256, 256, 0, stream>>>(feature, feat);
    }

    // 2) b1 = sigmoid(Wb1 @ X + bb1[:,None])   X[k][c] = inp[k*80+c]
    //    M=1024 N=80 K=300 ; NT=5 covers all of N per wave
    wmma_gemm_f32<2,1,true,false,5><<<dim3(1, DE1/16), 32, 0, stream>>>(
        Wb1, inp, bb1, b1, DE, /*rsA*/DE, /*csA*/1, /*rsB*/CC, /*csB*/1, /*ldd*/CC, 1.0f);
    // 3) b2
    wmma_gemm_f32<2,1,true,false,5><<<dim3(1, DE1/16), 32, 0, stream>>>(
        Wb2, inp, bb2, b2, DE, DE, 1, CC, 1, CC, 1.0f);

    // 4) Araw = (b1^T @ b2) / 80    A_op[m,k]=b1[k,m] -> rsA=1,csA=80
    wmma_gemm_f32<0,0,false,false,5><<<dim3(1, CC/16), 32, 0, stream>>>(
        b1, b2, nullptr, Araw, DE1, /*rsA*/1, /*csA*/CC, /*rsB*/CC, /*csB*/1, CC,
        1.0f / (float)CC);

    // 5) A_hat + L_A_loss
    ahat_kernel<<<1, 256, 0, stream>>>(Araw, Ahat, out + BB * CC);

    // 6) t1 = inp @ Wg1   M=80 N=1024 K=300
    wmma_gemm_f32<0,0,true,false,2><<<dim3(DE1/32, CC/16), 32, 0, stream>>>(
        inp, Wg1, nullptr, t1, DE, DE, 1, DE1, 1, DE1, 1.0f);

    // 7) h = leaky_relu(A_hat @ t1, 0.2)   M=80 N=1024 K=80
    wmma_gemm_f32<0,2,true,false,2><<<dim3(DE1/32, CC/16), 32, 0, stream>>>(
        Ahat, t1, nullptr, h, CC, CC, 1, DE1, 1, DE1, 1.0f);

    // 8) t2 = h @ Wg2   M=80 N=2048 K=1024
    wmma_gemm_f32<0,0,true,false,2><<<dim3(FDIM/32, CC/16), 32, 0, stream>>>(
        h, Wg2, nullptr, t2, DE1, DE1, 1, FDIM, 1, FDIM, 1.0f);

    // 9) x = A_hat @ t2   M=80 N=2048 K=80
    wmma_gemm_f32<0,0,true,false,2><<<dim3(FDIM/32, CC/16), 32, 0, stream>>>(
        Ahat, t2, nullptr, x, CC, CC, 1, FDIM, 1, FDIM, 1.0f);

    // 10) img = feat @ Wimg^T + bimg   M=64 N=4000 K=2048 ; B[k,n]=Wimg[n*2048+k]
    wmma_gemm_f32<1,0,true,true,2><<<dim3(JJ/32, BB/16), 32, 0, stream>>>(
        feat, Wimg, bimg, img, FDIM, FDIM, 1, /*rsB*/1, /*csB*/FDIM, JJ, 1.0f);

    // 11) cls = x @ Wcls^T + bcls   M=80 N=4000 K=2048
    wmma_gemm_f32<1,0,true,true,2><<<dim3(JJ/32, CC/16), 32, 0, stream>>>(
        x, Wcls, bcls, cls, FDIM, FDIM, 1, 1, FDIM, JJ, 1.0f);

    // 12) pooled[b,c,k] = sum_s img[b,k*100+s] * cls[c,k*100+s]
    {
        int n = BB * CC * KK;
        mfb_pool_kernel<<<(n + 255) / 256, 256, 0, stream>>>(img, cls, pooled);
    }

    // 13) out = pooled @ Wml^T + bml   M=64 N=80 K=3200 ; B[k,n]=Wml[n*3200+k]
    wmma_gemm_f32<1,0,true,true,5><<<dim3(1, BB/16), 32, 0, stream>>>(
        pooled, Wml, bml, out, CC * KK, /*rsA*/CC * KK, 1, /*rsB*/1, /*csB*/CC * KK,
        CC, 1.0f);

    (void)in_sizes; (void)n_in; (void)out_size; (void)ws_size;
}